// PointNetSetAbstraction_70970039599961
// MI455X (gfx1250) — compile-verified
//
#include <hip/hip_runtime.h>
#include <hip/hip_bf16.h>
#include <stdint.h>

typedef float v2f __attribute__((ext_vector_type(2)));
typedef float v8f __attribute__((ext_vector_type(8)));

#define BB 16
#define NN 8192
#define DD 9
#define SS 1024            // NPOINT
#define KK 32              // NSAMPLE
#define KS (KK*SS)         // 32768
#define PP ((size_t)BB*KS) // 524288 positions
#define R2F 0.04f
#define EPSF 1e-5f

// monotone float -> u32 total-order transform (handles negatives)
__device__ __forceinline__ uint32_t ford(float f){
  uint32_t u = __float_as_uint(f);
  return (u & 0x80000000u) ? ~u : (u | 0x80000000u);
}

// ---------------------------------------------------------------------------
// 1) Farthest point sampling: one workgroup per batch, 1024 threads x 8 pts.
//    argmax with tie-break to LOWEST index via packed (distBits<<32)|~idx max.
// ---------------------------------------------------------------------------
__global__ __launch_bounds__(1024)
void fps_kernel(const float* __restrict__ xyz, int* __restrict__ cent){
  const int b = blockIdx.x;
  const int t = threadIdx.x;
  __shared__ unsigned long long red[1024];
  __shared__ float shc[3];
  float px[8], py[8], pz[8], dist[8];
  const float* base = xyz + (size_t)b * NN * 3;
#pragma unroll
  for (int j = 0; j < 8; ++j){
    int n = j*1024 + t;
    px[j] = base[n*3+0]; py[j] = base[n*3+1]; pz[j] = base[n*3+2];
    dist[j] = 1e10f;
  }
  int far = 0;
  for (int i = 0; i < SS; ++i){
    if (t == 0) cent[b*SS + i] = far;
    if (t == (far & 1023)){
      int jj = far >> 10;
      float cx = px[0], cy = py[0], cz = pz[0];
#pragma unroll
      for (int j = 1; j < 8; ++j) if (jj == j){ cx = px[j]; cy = py[j]; cz = pz[j]; }
      shc[0] = cx; shc[1] = cy; shc[2] = cz;
    }
    __syncthreads();
    const float cx = shc[0], cy = shc[1], cz = shc[2];
    unsigned long long best = 0ull;
#pragma unroll
    for (int j = 0; j < 8; ++j){
      float dx = px[j]-cx, dy = py[j]-cy, dz = pz[j]-cz;
      float d = dx*dx + dy*dy + dz*dz;
      d = fminf(dist[j], d); dist[j] = d;
      unsigned long long key = ((unsigned long long)__float_as_uint(d) << 32)
                             | (uint32_t)(~(uint32_t)(j*1024 + t));
      best = (key > best) ? key : best;
    }
    red[t] = best;
    __syncthreads();
    for (int sdx = 512; sdx > 0; sdx >>= 1){
      if (t < sdx){ unsigned long long o = red[t+sdx]; if (o > red[t]) red[t] = o; }
      __syncthreads();
    }
    far = (int)(~(uint32_t)red[0]);
    __syncthreads();
  }
}

// ---------------------------------------------------------------------------
// 2) new_xyz gather -> first part of d_out (B,S,3)
// ---------------------------------------------------------------------------
__global__ void newxyz_kernel(const float* __restrict__ xyz,
                              const int* __restrict__ cent,
                              float* __restrict__ out){
  int i = blockIdx.x*256 + threadIdx.x;           // over B*S
  if (i >= BB*SS) return;
  int b = i >> 10; int c = cent[i];
  const float* p = xyz + ((size_t)b*NN + c)*3;
  out[i*3+0]=p[0]; out[i*3+1]=p[1]; out[i*3+2]=p[2];
}

// ---------------------------------------------------------------------------
// 3) 32-nearest selection per query. One workgroup (256 thr) per (b,s).
//    Distance row (8192 f32 = 32KB) in LDS, 32 argmin+exclude iterations.
//    Out-of-radius entries replaced by the nearest index (iteration 0).
// ---------------------------------------------------------------------------
__global__ __launch_bounds__(256)
void query_kernel(const float* __restrict__ xyz, const int* __restrict__ cent,
                  int* __restrict__ idxb){
  __shared__ float ld[NN];                        // 32 KB
  __shared__ unsigned long long red[256];
  __shared__ int sout[KK];
  const int bs = blockIdx.x;
  const int b  = bs >> 10;
  const int t  = threadIdx.x;
  const float* base = xyz + (size_t)b*NN*3;
  const int qi = cent[bs];
  const float qx = base[qi*3+0], qy = base[qi*3+1], qz = base[qi*3+2];
  const float q2 = qx*qx+qy*qy+qz*qz;
#pragma unroll
  for (int j = 0; j < NN/256; ++j){
    int n = j*256 + t;
    float x = base[n*3], y = base[n*3+1], z = base[n*3+2];
    ld[n] = q2 + (x*x+y*y+z*z) - 2.0f*(qx*x+qy*y+qz*z);
  }
  __syncthreads();
  const uint32_t keyR = ford(R2F);
  for (int it = 0; it < KK; ++it){
    unsigned long long best = ~0ull;
#pragma unroll
    for (int j = 0; j < NN/256; ++j){
      int n = j*256 + t;
      unsigned long long key = ((unsigned long long)ford(ld[n]) << 32) | (uint32_t)n;
      best = (key < best) ? key : best;
    }
    red[t] = best;
    __syncthreads();
    for (int sdx = 128; sdx > 0; sdx >>= 1){
      if (t < sdx){ unsigned long long o = red[t+sdx]; if (o < red[t]) red[t] = o; }
      __syncthreads();
    }
    unsigned long long win = red[0];
    if (t == 0){
      uint32_t nw = (uint32_t)win;
      int sel;
      if (it == 0) sel = (int)nw;
      else sel = ((uint32_t)(win >> 32) <= keyR) ? (int)nw : sout[0];
      sout[it] = sel;
      ld[nw] = 3.0e38f;                           // exclude winner
    }
    __syncthreads();
  }
  if (t < KK) idxb[bs*KK + t] = sout[t];
}

// ---------------------------------------------------------------------------
// 4) Gather points by neighbor idx into channel-major X0[c][pos],
//    pos = b*KS + k*S + s
// ---------------------------------------------------------------------------
__global__ void gather_kernel(const float* __restrict__ points,
                              const int* __restrict__ idxb,
                              float* __restrict__ X0){
  size_t pos = (size_t)blockIdx.x*256 + threadIdx.x;
  if (pos >= PP) return;
  int s = (int)(pos & (SS-1));
  int k = (int)((pos >> 10) & (KK-1));
  int b = (int)(pos >> 15);
  int pt = idxb[(b*SS + s)*KK + k];
  const float* src = points + ((size_t)b*NN + pt)*DD;
#pragma unroll
  for (int c = 0; c < DD; ++c) X0[(size_t)c*PP + pos] = src[c];
}

__global__ void zero_stats_kernel(float* stats){
  int t = blockIdx.x*256 + threadIdx.x;
  if (t < 6*128) stats[t] = 0.f;
}

// ---------------------------------------------------------------------------
// 5) Layer0: cin=9, scalar FMA (K too small for WMMA). Emits stats to LDS,
//    then one global atomic per channel per block.
// ---------------------------------------------------------------------------
__global__ __launch_bounds__(256)
void l0_kernel(const float* __restrict__ X0, const float* __restrict__ W0,
               const float* __restrict__ b0, float* __restrict__ Y,
               float* __restrict__ stats){
  __shared__ float lw[64*9];
  __shared__ float lb[64];
  __shared__ float ssum[64], ssq[64];
  int t = threadIdx.x;
  for (int i = t; i < 64*9; i += 256) lw[i] = W0[i];
  if (t < 64){ lb[t] = b0[t]; ssum[t] = 0.f; ssq[t] = 0.f; }
  __syncthreads();
  size_t pos = (size_t)blockIdx.x*256 + t;
  float x[DD];
#pragma unroll
  for (int c = 0; c < DD; ++c) x[c] = X0[(size_t)c*PP + pos];
#pragma unroll 4
  for (int o = 0; o < 64; ++o){
    float acc = lb[o];
#pragma unroll
    for (int c = 0; c < DD; ++c) acc = fmaf(lw[o*DD+c], x[c], acc);
    Y[(size_t)o*PP + pos] = acc;
    atomicAdd(&ssum[o], acc);
    atomicAdd(&ssq[o], acc*acc);
  }
  __syncthreads();
  if (t < 64){
    atomicAdd(&stats[t],       ssum[t]);
    atomicAdd(&stats[128 + t], ssq[t]);
  }
}

// ---------------------------------------------------------------------------
// 6) BatchNorm(+bias already applied) + ReLU, in place. mean/var over P.
// ---------------------------------------------------------------------------
__global__ void bn_kernel(float* __restrict__ Y, const float* __restrict__ stats,
                          const float* __restrict__ gamma,
                          const float* __restrict__ beta){
  int o = blockIdx.y;
  size_t p = (size_t)blockIdx.x*256 + threadIdx.x;
  const float invP = 1.0f/(float)PP;
  float m = stats[o] * invP;
  float v = stats[128+o] * invP - m*m;
  float rs = rsqrtf(fmaxf(v, 0.f) + EPSF);
  float g = gamma[o], be = beta[o];
  float y = Y[(size_t)o*PP + p];
  y = (y - m)*rs*g + be;
  Y[(size_t)o*PP + p] = fmaxf(y, 0.f);
}

// ---------------------------------------------------------------------------
// 7) WMMA GEMM: Y(cout x P) = W(cout x 64) * X(64 x P) + bias, fp32 WMMA.
//    Wave owns 16 positions and all COUT/16 row tiles; K=64 in 16 steps of
//    v_wmma_f32_16x16x4_f32. W staged in LDS. BN stats via LDS atomics.
//    A: lane(M=lane&15), elem r -> K = 2*(lane>>4)+r ; B symmetric with N.
//    C/D: VGPR v -> M = v + 8*(lane>>4), N = lane&15.
// ---------------------------------------------------------------------------
template<int COUT>
__global__ __launch_bounds__(128)
void gemm_wmma_kernel(const float* __restrict__ X, const float* __restrict__ W,
                      const float* __restrict__ bias, float* __restrict__ Y,
                      float* __restrict__ stats){
  constexpr int NT = COUT/16;
  __shared__ float lw[COUT*64];
  __shared__ float lb[COUT];
  __shared__ float ssum[COUT], ssq[COUT];
  const int t = threadIdx.x;
  for (int i = t; i < COUT*64; i += 128) lw[i] = W[i];
  if (t < COUT){ lb[t] = bias[t]; ssum[t] = 0.f; ssq[t] = 0.f; }
  __syncthreads();

  const int wave = t >> 5;
  const int lane = t & 31;
  const int h    = lane >> 4;
  const int n    = lane & 15;
  const size_t posb = ((size_t)blockIdx.x*4 + wave) * 16;

  v8f acc[NT];
#pragma unroll
  for (int rt = 0; rt < NT; ++rt)
#pragma unroll
    for (int e = 0; e < 8; ++e) acc[rt][e] = 0.f;

#pragma unroll
  for (int kk = 0; kk < 16; ++kk){
    const int c0 = kk*4 + 2*h;                 // this lane-half's K pair
    v2f bfrag;
    bfrag.x = X[(size_t)(c0+0)*PP + posb + n];
    bfrag.y = X[(size_t)(c0+1)*PP + posb + n];
#pragma unroll
    for (int rt = 0; rt < NT; ++rt){
      v2f afrag;
      afrag.x = lw[(rt*16 + n)*64 + c0 + 0];
      afrag.y = lw[(rt*16 + n)*64 + c0 + 1];
      acc[rt] = __builtin_amdgcn_wmma_f32_16x16x4_f32(
          false, afrag, false, bfrag, (short)0, acc[rt], false, false);
    }
  }

#pragma unroll
  for (int rt = 0; rt < NT; ++rt){
#pragma unroll
    for (int v = 0; v < 8; ++v){
      int o = rt*16 + v + 8*h;
      float y = acc[rt][v] + lb[o];
      Y[(size_t)o*PP + posb + n] = y;
      atomicAdd(&ssum[o], y);
      atomicAdd(&ssq[o], y*y);
    }
  }
  __syncthreads();
  if (t < COUT){
    atomicAdd(&stats[t],       ssum[t]);
    atomicAdd(&stats[128 + t], ssq[t]);
  }
}

// ---------------------------------------------------------------------------
// 8) Fused BN + ReLU + max over k for final layer -> d_out feat (B,S,128)
// ---------------------------------------------------------------------------
__global__ void maxbn_kernel(const float* __restrict__ Y3,
                             const float* __restrict__ stats,
                             const float* __restrict__ gamma,
                             const float* __restrict__ beta,
                             float* __restrict__ out){
  int o = blockIdx.y;
  int i = blockIdx.x*256 + threadIdx.x;          // bs
  int b = i >> 10, s = i & (SS-1);
  const float invP = 1.0f/(float)PP;
  float m = stats[o] * invP;
  float v = stats[128+o] * invP - m*m;
  float rs = rsqrtf(fmaxf(v, 0.f) + EPSF);
  float g = gamma[o], be = beta[o];
  const float* src = Y3 + (size_t)o*PP + (size_t)b*KS + s;
  float mx = -3.4e38f;
#pragma unroll
  for (int k = 0; k < KK; ++k){
    float y = src[(size_t)k*SS];
    y = fmaxf((y - m)*rs*g + be, 0.f);
    mx = fmaxf(mx, y);
  }
  out[(size_t)i*128 + o] = mx;
}

// ---------------------------------------------------------------------------
extern "C" void kernel_launch(void* const* d_in, const int* in_sizes, int n_in,
                              void* d_out, int out_size, void* d_ws, size_t ws_size,
                              hipStream_t stream) {
  const float* xyz    = (const float*)d_in[0];
  const float* points = (const float*)d_in[1];
  const float* W0 = (const float*)d_in[2];
  const float* b0 = (const float*)d_in[3];
  const float* g0 = (const float*)d_in[4];
  const float* be0= (const float*)d_in[5];
  const float* W1 = (const float*)d_in[6];
  const float* b1 = (const float*)d_in[7];
  const float* g1 = (const float*)d_in[8];
  const float* be1= (const float*)d_in[9];
  const float* W2 = (const float*)d_in[10];
  const float* b2 = (const float*)d_in[11];
  const float* g2 = (const float*)d_in[12];
  const float* be2= (const float*)d_in[13];

  char* ws = (char*)d_ws;
  size_t off = 0;
  auto alloc = [&](size_t bytes)->void*{
    void* p = ws + off;
    off = (off + bytes + 255) & ~(size_t)255;
    return p;
  };
  int*   cent  = (int*)  alloc((size_t)BB*SS*4);
  int*   idxb  = (int*)  alloc((size_t)BB*SS*KK*4);
  float* stats = (float*)alloc(6*128*4);
  float* X0    = (float*)alloc((size_t)DD *PP*4);
  float* Y1    = (float*)alloc((size_t)64 *PP*4);
  float* Y2    = (float*)alloc((size_t)64 *PP*4);
  float* Y3    = (float*)alloc((size_t)128*PP*4);
  float* out   = (float*)d_out;

  fps_kernel   <<<BB, 1024, 0, stream>>>(xyz, cent);
  newxyz_kernel<<<(BB*SS+255)/256, 256, 0, stream>>>(xyz, cent, out);
  query_kernel <<<BB*SS, 256, 0, stream>>>(xyz, cent, idxb);
  gather_kernel<<<(int)(PP/256), 256, 0, stream>>>(points, idxb, X0);
  zero_stats_kernel<<<3, 256, 0, stream>>>(stats);

  l0_kernel<<<(int)(PP/256), 256, 0, stream>>>(X0, W0, b0, Y1, stats + 0);
  bn_kernel<<<dim3((unsigned)(PP/256), 64), 256, 0, stream>>>(Y1, stats + 0, g0, be0);

  gemm_wmma_kernel<64> <<<(int)(PP/64), 128, 0, stream>>>(Y1, W1, b1, Y2, stats + 256);
  bn_kernel<<<dim3((unsigned)(PP/256), 64), 256, 0, stream>>>(Y2, stats + 256, g1, be1);

  gemm_wmma_kernel<128><<<(int)(PP/64), 128, 0, stream>>>(Y2, W2, b2, Y3, stats + 512);
  maxbn_kernel<<<dim3(BB*SS/256, 128), 256, 0, stream>>>(Y3, stats + 512, g2, be2,
                                                         out + (size_t)BB*SS*3);
}